// Refiner_89962384982056
// MI455X (gfx1250) — compile-verified
//
#include <hip/hip_runtime.h>
#include <cstdint>

#define BLOCK   128
#define KDIM    64
#define TSTEPS  10

// psi = C_TH * sin(theta_rad) + C_R * r ;  C_TH = 2*pi*F0*D_EL/C = pi (exact)
#define C_TH      3.14159265358979323846f
#define C_R      (-1.2566370614359172e-3f)   // -2*pi*(2*DF/C) = -2*pi*2e-4
#define DEG2RAD   0.017453292519943295f
#define TH_HALF_  60.0f
#define RMAX_     2000.0f
#define A_MIN_    1e-5f
#define A_MAX_    0.05f
#define L_MIN_    1e-3f
#define L_MAX_    1.0f
#define STEP_CLIP_ 0.25f

__device__ __forceinline__ float sigmoidf_(float x) { return 1.0f / (1.0f + expf(-x)); }

__device__ __forceinline__ void async_b32(uint32_t lds_byte_addr, const void* gaddr) {
    asm volatile("global_load_async_to_lds_b32 %0, %1, off"
                 :: "v"(lds_byte_addr), "v"((uint64_t)(size_t)gaddr) : "memory");
}

__global__ __launch_bounds__(BLOCK, 1) void fda_refine_kernel(
    const float* __restrict__ z_real, const float* __restrict__ z_imag,
    const float* __restrict__ theta0, const float* __restrict__ r0,
    const float* __restrict__ alpha_raw, const float* __restrict__ lambda_raw,
    float* __restrict__ out_theta, float* __restrict__ out_r, float* __restrict__ out_J,
    int B)
{
    // z tile for this block, transposed: zz[k*BLOCK + local_b] = (re, im). 64 KB.
    __shared__ float2 zz[KDIM * BLOCK];

    const int tid = threadIdx.x;
    const long long b0    = (long long)blockIdx.x * BLOCK;
    const long long gbase = b0 * KDIM;
    const long long total = (long long)B * KDIM;
    const long long b     = b0 + tid;

    const float* zr = z_real + gbase;
    const float* zi = z_imag + gbase;

    // ---- stage z into LDS (transposed) via CDNA5 async global->LDS DMA ----
    // flat = c*128 + tid  =>  k = tid&63 is per-thread constant, bl = 2c + (tid>>6).
    // Global side: lanes read consecutive floats (fully coalesced).
    // LDS side: per-lane scatter address performs the [b][k] -> [k][b] transpose.
    {
        const int k0  = tid & (KDIM - 1);
        const int bl0 = tid >> 6;
        uint32_t laddr = (uint32_t)(size_t)(&zz[k0 * BLOCK + bl0]);
        const float* gr = zr + tid;
        const float* gi = zi + tid;
        if (gbase + (long long)(KDIM * BLOCK) <= total) {
            // full tile: no per-iteration exec masking
#pragma unroll 4
            for (int c = 0; c < KDIM; ++c) {
                async_b32(laddr,      gr);
                async_b32(laddr + 4u, gi);
                laddr += 16u;          // bl advances by 2 float2's
                gr += BLOCK;
                gi += BLOCK;
            }
        } else {
            for (int c = 0; c < KDIM; ++c) {
                long long flat = (long long)c * BLOCK + tid;
                if (gbase + flat < total) {
                    async_b32(laddr,      gr);
                    async_b32(laddr + 4u, gi);
                }
                laddr += 16u;
                gr += BLOCK;
                gi += BLOCK;
            }
        }
    }

    // ---- overlap with DMA: schedule + u-init (independent of LDS contents) ----
    float alf[TSTEPS], lam[TSTEPS];
#pragma unroll
    for (int t = 0; t < TSTEPS; ++t) {
        alf[t] = A_MIN_ + (A_MAX_ - A_MIN_) * sigmoidf_(alpha_raw[t]);
        lam[t] = L_MIN_ + (L_MAX_ - L_MIN_) * sigmoidf_(lambda_raw[t]);
    }

    const bool valid = (b < (long long)B);
    float th0 = valid ? theta0[b] : 0.0f;
    float rr0 = valid ? r0[b]     : 0.0f;
    float y  = fminf(fmaxf(th0 * (1.0f / TH_HALF_), -1.0f + 1e-6f), 1.0f - 1e-6f);
    float u0 = 0.5f * (log1pf(y) - log1pf(-y));
    float p  = fminf(fmaxf(rr0 * (1.0f / RMAX_), 1e-6f), 1.0f - 1e-6f);
    float u1 = logf(p) - log1pf(-p);

    // ---- DMA completion + cross-wave visibility ----
#if defined(__has_builtin) && __has_builtin(__builtin_amdgcn_s_wait_asynccnt)
    __builtin_amdgcn_s_wait_asynccnt(0);
#else
    asm volatile("s_wait_asynccnt 0" ::: "memory");
#endif
    __syncthreads();

    if (!valid) return;

    // ---- ||z||^2 (constant over iterations) ----
    float zn = 0.0f;
#pragma unroll 8
    for (int k = 0; k < KDIM; ++k) {
        float2 v = zz[k * BLOCK + tid];
        zn = fmaf(v.x, v.x, zn);
        zn = fmaf(v.y, v.y, zn);
    }
    const float invden = 1.0f / fmaf(64.0f, zn, 6.4e-11f);  // J = |S|^2 * invden

    // ---- T refinement steps: closed-form gradient via S, S1 recurrences ----
#pragma unroll 1
    for (int t = 0; t < TSTEPS; ++t) {
        float t0  = tanhf(u0);
        float thr = (TH_HALF_ * DEG2RAD) * t0;            // theta in radians
        float st, ct;  sincosf(thr, &st, &ct);
        float sg  = sigmoidf_(u1);
        float r   = RMAX_ * sg;
        float psi = fmaf(C_TH, st, C_R * r);
        float sp, cp;  sincosf(psi, &sp, &cp);
        float dr_ = cp, di_ = -sp;                        // e^{-i psi}

        float wr = 1.0f, wi = 0.0f;                       // w_k = e^{-i k psi}
        float Sre = 0.0f, Sim = 0.0f, S1re = 0.0f, S1im = 0.0f, kf = 0.0f;
#pragma unroll 8
        for (int k = 0; k < KDIM; ++k) {
            float2 v = zz[k * BLOCK + tid];
            float tr = v.x * wr - v.y * wi;               // z_k * w_k
            float ti = v.x * wi + v.y * wr;
            Sre += tr;  Sim += ti;
            S1re = fmaf(kf, tr, S1re);
            S1im = fmaf(kf, ti, S1im);
            float nwr = wr * dr_ - wi * di_;              // w *= e^{-i psi}
            float nwi = fmaf(wr, di_, wi * dr_);
            wr = nwr;  wi = nwi;
            kf += 1.0f;
        }

        // d(-J)/dpsi = -2*Im(conj(S)*S1)/den
        float gpsi  = -2.0f * (Sre * S1im - Sim * S1re) * invden;
        float dpdu0 = (C_TH * DEG2RAD * TH_HALF_) * ct * (1.0f - t0 * t0);
        float dpdu1 = (C_R * RMAX_) * sg * (1.0f - sg);
        float g0 = gpsi * dpdu0;
        float g1 = gpsi * dpdu1;
        if (!isfinite(g0)) g0 = 0.0f;
        if (!isfinite(g1)) g1 = 0.0f;

        float d0 = fmaxf(fabsf(g0), 1e-6f) + lam[t];
        float d1 = fmaxf(fabsf(g1), 1e-6f) + lam[t];
        if (!isfinite(d0)) d0 = 1.0f;
        if (!isfinite(d1)) d1 = 1.0f;
        d0 = fmaxf(d0, 1e-6f);
        d1 = fmaxf(d1, 1e-6f);

        float s0 = fminf(fmaxf(alf[t] * g0 / d0, -STEP_CLIP_), STEP_CLIP_);
        float s1 = fminf(fmaxf(alf[t] * g1 / d1, -STEP_CLIP_), STEP_CLIP_);
        u0 -= s0;
        u1 -= s1;
    }

    // ---- final outputs ----
    float t0    = tanhf(u0);
    float theta = TH_HALF_ * t0;
    float sg    = sigmoidf_(u1);
    float r     = RMAX_ * sg;
    float st    = sinf(theta * DEG2RAD);
    float psi   = fmaf(C_TH, st, C_R * r);
    float sp, cp;  sincosf(psi, &sp, &cp);
    float dr_ = cp, di_ = -sp;
    float wr = 1.0f, wi = 0.0f, Sre = 0.0f, Sim = 0.0f;
#pragma unroll 8
    for (int k = 0; k < KDIM; ++k) {
        float2 v = zz[k * BLOCK + tid];
        Sre = fmaf(v.x, wr, Sre);  Sre = fmaf(-v.y, wi, Sre);
        Sim = fmaf(v.x, wi, Sim);  Sim = fmaf(v.y, wr, Sim);
        float nwr = wr * dr_ - wi * di_;
        float nwi = fmaf(wr, di_, wi * dr_);
        wr = nwr;  wi = nwi;
    }
    float J = fmaf(Sre, Sre, Sim * Sim) * invden;

    out_theta[b] = theta;
    out_r[b]     = r;
    out_J[b]     = J;
}

extern "C" void kernel_launch(void* const* d_in, const int* in_sizes, int n_in,
                              void* d_out, int out_size, void* d_ws, size_t ws_size,
                              hipStream_t stream) {
    const float* z_real = (const float*)d_in[0];
    const float* z_imag = (const float*)d_in[1];
    const float* theta0 = (const float*)d_in[2];
    const float* r0     = (const float*)d_in[3];
    const float* a_raw  = (const float*)d_in[4];
    const float* l_raw  = (const float*)d_in[5];
    const int B = in_sizes[2];   // theta0 has B elements

    float* out = (float*)d_out;  // [theta(B) | r(B) | J(B)]
    dim3 grid((unsigned)((B + BLOCK - 1) / BLOCK));
    fda_refine_kernel<<<grid, dim3(BLOCK), 0, stream>>>(
        z_real, z_imag, theta0, r0, a_raw, l_raw,
        out, out + B, out + 2 * B, B);
}